// MultiresLayer_18064632447417
// MI455X (gfx1250) — compile-verified
//
#include <hip/hip_runtime.h>
#include <math.h>

typedef __attribute__((ext_vector_type(2))) float        v2f;
typedef __attribute__((ext_vector_type(4))) float        v4f;
typedef __attribute__((ext_vector_type(8))) float        v8f;
typedef __attribute__((ext_vector_type(4))) unsigned int v4u;
typedef __attribute__((ext_vector_type(4))) int          v4i;
typedef __attribute__((ext_vector_type(8))) int          v8i;

#define B_      16
#define C_      256
#define L_      8192
#define K_      4
#define DEPTH_  12
#define THREADS 256   // 8 waves of 32 (wave32)

// One workgroup per (b,c) row. Whole 32KB row lives in LDS; the 12-level
// dilated cascade ping-pongs between two LDS planes, y accumulates in a third.
// Conv levels run on the matrix pipe via V_WMMA_F32_16X16X4_F32 (K==taps).
__global__ __launch_bounds__(THREADS) void multires_wmma_kernel(
    const float* __restrict__ x,  const float* __restrict__ h0,
    const float* __restrict__ h1, const float* __restrict__ w,
    float* __restrict__ out)
{
    __shared__ float sA[L_];   // res_lo ping (starts as x, loaded by TDM)
    __shared__ float sB[L_];   // res_lo pong
    __shared__ float sY[L_];   // y accumulator

    const int row  = blockIdx.x;        // b*C + c
    const int c    = row & (C_ - 1);
    const int tid  = threadIdx.x;
    const int lane = tid & 31;
    const int wv   = tid >> 5;          // wave id 0..7

    const float* xrow = x   + (size_t)row * L_;
    float*       orow = out + (size_t)row * L_;

    // ---- Stage x row into sA with the Tensor Data Mover (async DMA) ----
    if (wv == 0) {
        unsigned long long ga = (unsigned long long)xrow;
        unsigned lds_off = (unsigned)(unsigned long long)(void*)sA; // LDS byte offset
        v4u g0;
        g0[0] = 1u;                                   // count=1, user mode
        g0[1] = lds_off;                              // lds_addr
        g0[2] = (unsigned)ga;                         // global_addr[31:0]
        g0[3] = (unsigned)((ga >> 32) & 0x01FFFFFFull) | 0x80000000u; // addr hi | type=2
        v8i g1;
        g1[0] = (int)(2u << 16);                      // data_size = 4B
        g1[1] = (int)(((unsigned)L_ & 0xFFFFu) << 16);// tensor_dim0[15:0]
        g1[2] = (int)(((unsigned)L_ >> 16) | (1u << 16)); // dim0 hi | tensor_dim1=1 lo
        g1[3] = (int)((unsigned)L_ << 16);            // tensor_dim1 hi=0 | tile_dim0=8192
        g1[4] = 1;                                    // tile_dim1=1, tile_dim2=0
        g1[5] = L_;                                   // tensor_dim0_stride lo32
        g1[6] = 0;
        g1[7] = 0;
        v4i g2 = {0, 0, 0, 0};
        v4i g3 = {0, 0, 0, 0};
        v8i g4 = {0, 0, 0, 0, 0, 0, 0, 0};            // 6-arg toolchain: extra group
        __builtin_amdgcn_tensor_load_to_lds(g0, g1, g2, g3, g4, 0);
        __builtin_amdgcn_s_wait_tensorcnt(0);
    }
    __syncthreads();

    // ---- y = w[c, DEPTH+1] * x ----
    const float w_last = w[c * (DEPTH_ + 2) + (DEPTH_ + 1)];
    #pragma unroll
    for (int j = 0; j < 8; ++j) {
        int idx = (tid + THREADS * j) * 4;
        v4f xv = *(const v4f*)(sA + idx);
        *(v4f*)(sY + idx) = xv * w_last;
    }
    __syncthreads();

    // per-channel filters (uniform across block -> scalar loads)
    float f0[4], f1[4];
    #pragma unroll
    for (int k = 0; k < 4; ++k) { f0[k] = h0[c * 4 + k]; f1[k] = h1[c * 4 + k]; }

    float* ping = sA;
    float* pong = sB;

    const int l16   = lane & 15;    // A: row m   | B/D: column n
    const int hi    = lane >> 4;    // half-wave select
    const int kbase = hi * 2;       // K index held in .x (0 or 2), .y = +1

    for (int lev = 0; lev < DEPTH_; ++lev) {
        const int   d  = 1 << lev;
        const float wi = w[c * (DEPTH_ + 2) + (DEPTH_ - lev)];

        // B (4x16 f32): col0 = h0 reversed (-> res_lo), col1 = wi*h1 reversed (-> wi*res_hi)
        v2f bm;
        {
            float c0x = f0[3 - kbase],        c0y = f0[3 - (kbase + 1)];
            float c1x = wi * f1[3 - kbase],   c1y = wi * f1[3 - (kbase + 1)];
            bm.x = (l16 == 0) ? c0x : (l16 == 1) ? c1x : 0.0f;
            bm.y = (l16 == 0) ? c0y : (l16 == 1) ? c1y : 0.0f;
        }

        // 64 tiles of 16 outputs per wave: this wave covers [wv*1024, wv*1024+1024)
        for (int tt = 0; tt < 64; ++tt) {
            const int o0 = (wv * 64 + tt) * 16;
            // A[m,k] = ping[o0 - 3d + m + k*d], causal zero-pad for idx < 0.
            // Branchless guard: clamp index, unconditional ds_load, v_cndmask zero.
            const int i0 = o0 - 3 * d + l16 + kbase * d;
            const int i1 = i0 + d;
            const int j0 = (i0 > 0) ? i0 : 0;
            const int j1 = (i1 > 0) ? i1 : 0;
            float t0 = ping[j0];
            float t1 = ping[j1];
            v2f am;
            am.x = (i0 >= 0) ? t0 : 0.0f;
            am.y = (i1 >= 0) ? t1 : 0.0f;

            v8f cz = {};
            v8f dm = __builtin_amdgcn_wmma_f32_16x16x4_f32(
                false, am, false, bm, (short)0, cz, false, false);

            // D: lane = column; VGPR t holds row t (lanes 0-15) / t+8 (lanes 16-31)
            if (l16 == 0) {               // res_lo -> pong
                float* p = pong + o0 + hi * 8;
                #pragma unroll
                for (int t = 0; t < 8; ++t) p[t] = dm[t];
            } else if (l16 == 1) {        // wi*res_hi -> y (same lanes own same slots every level)
                float* p = sY + o0 + hi * 8;
                #pragma unroll
                for (int t = 0; t < 8; ++t) p[t] += dm[t];
            }
        }
        __syncthreads();
        float* tmp = ping; ping = pong; pong = tmp;
    }

    // ---- y += w[c,0] * res_lo; exact-erf GELU; coalesced store ----
    const float w0 = w[c * (DEPTH_ + 2) + 0];
    #pragma unroll
    for (int j = 0; j < 8; ++j) {
        int idx = (tid + THREADS * j) * 4;
        v4f rl = *(const v4f*)(ping + idx);
        v4f yv = *(const v4f*)(sY + idx);
        yv = yv + rl * w0;
        v4f g;
        #pragma unroll
        for (int e = 0; e < 4; ++e) {
            float v = yv[e];
            g[e] = 0.5f * v * (1.0f + erff(v * 0.70710678118654752f));
        }
        *(v4f*)(orow + idx) = g;
    }
}

extern "C" void kernel_launch(void* const* d_in, const int* in_sizes, int n_in,
                              void* d_out, int out_size, void* d_ws, size_t ws_size,
                              hipStream_t stream) {
    const float* x  = (const float*)d_in[0];
    const float* h0 = (const float*)d_in[1];
    const float* h1 = (const float*)d_in[2];
    const float* w  = (const float*)d_in[3];
    float* out = (float*)d_out;

    dim3 grid(B_ * C_);     // one workgroup per (b,c) row
    dim3 block(THREADS);
    multires_wmma_kernel<<<grid, block, 0, stream>>>(x, h0, h1, w, out);
}